// MRGNN_60851096650215
// MI455X (gfx1250) — compile-verified
//
#include <hip/hip_runtime.h>
#include <hip/hip_bf16.h>
#include <math.h>

// ---------------- problem constants (match reference) ----------------
#define R_REL 8
#define N_X 20000
#define N_GENE 4264
#define NT (N_X + N_GENE)          // 24264 nodes
#define HM 256
#define NUM_CLASSES 10
#define NE 400000
#define NE2 (2 * NE)
#define LDX 3613                    // 1613 + 2000
#define ACOL 1613
#define KX 2000
#define NTH ((size_t)NT * HM)       // 6,211,584

typedef float v2f __attribute__((ext_vector_type(2)));
typedef float v8f __attribute__((ext_vector_type(8)));

// ---------------- helpers ----------------
__device__ __forceinline__ float wave_reduce_sum(float v) {
  #pragma unroll
  for (int off = 16; off > 0; off >>= 1) v += __shfl_xor(v, off, 32);
  return v;
}

__device__ __forceinline__ float block_reduce_sum(float v, float* sdata) {
  int tid = threadIdx.x;
  sdata[tid] = v;
  __syncthreads();
  for (int s = blockDim.x >> 1; s > 0; s >>= 1) {
    if (tid < s) sdata[tid] += sdata[tid + s];
    __syncthreads();
  }
  float r = sdata[0];
  __syncthreads();
  return r;
}

// ---------------- WMMA f32 GEMM: C[M x 256] = A[M x K] @ W[K x 256] + bias ----------------
#define BLK_M 128
#define BLK_N 64
#define BLK_K 16

__global__ __launch_bounds__(256) void gemm_bias_kernel(
    const float* __restrict__ A, int lda, int acol0, int M, int K,
    const float* __restrict__ W, const float* __restrict__ bias,
    float* __restrict__ C)
{
  __shared__ float As[BLK_M][BLK_K + 1];   // [m][k]
  __shared__ float Ws[BLK_K][BLK_N + 4];   // [k][n]

  const int tid  = threadIdx.x;
  const int lane = tid & 31;
  const int wid  = tid >> 5;
  const int wm   = wid >> 1;               // 0..3 : 32-row subtile
  const int wn   = wid & 1;                // 0..1 : 32-col subtile
  const int bm   = blockIdx.x * BLK_M;
  const int bn   = blockIdx.y * BLK_N;

  v8f c[2][2] = {};

  const int kTiles = (K + BLK_K - 1) / BLK_K;
  for (int kt = 0; kt < kTiles; ++kt) {
    const int k0 = kt * BLK_K;
    // stage A tile (128x16): 8 elems/thread
    #pragma unroll
    for (int i = 0; i < 8; ++i) {
      int idx = tid + i * 256;
      int m = idx >> 4, k = idx & 15;
      int gr = bm + m, gk = k0 + k;
      As[m][k] = (gr < M && gk < K) ? A[(size_t)gr * lda + acol0 + gk] : 0.0f;
    }
    // stage W tile (16x64): 4 elems/thread
    #pragma unroll
    for (int i = 0; i < 4; ++i) {
      int idx = tid + i * 256;
      int kk = idx >> 6, n = idx & 63;
      int gk = k0 + kk;
      Ws[kk][n] = (gk < K) ? W[(size_t)gk * HM + bn + n] : 0.0f;
    }
    __syncthreads();

    #pragma unroll
    for (int ks = 0; ks < 4; ++ks) {
      // A 16x4 f32 layout: lanes 0-15 -> K={k,k+1}, lanes 16-31 -> K={k+2,k+3}
      const int kb = ks * 4 + ((lane >> 4) << 1);
      v2f a[2], b[2];
      #pragma unroll
      for (int mi = 0; mi < 2; ++mi) {
        int ml = wm * 32 + mi * 16 + (lane & 15);
        a[mi].x = As[ml][kb];
        a[mi].y = As[ml][kb + 1];
      }
      #pragma unroll
      for (int ni = 0; ni < 2; ++ni) {
        int nl = wn * 32 + ni * 16 + (lane & 15);
        b[ni].x = Ws[kb][nl];
        b[ni].y = Ws[kb + 1][nl];
      }
      #pragma unroll
      for (int mi = 0; mi < 2; ++mi)
        #pragma unroll
        for (int ni = 0; ni < 2; ++ni)
          c[mi][ni] = __builtin_amdgcn_wmma_f32_16x16x4_f32(
              false, a[mi], false, b[ni], (short)0, c[mi][ni], false, false);
    }
    __syncthreads();
  }

  // C/D layout: vgpr v -> rows {v, v+8}; lanes 0-15 row base 0, lanes 16-31 row base 8
  #pragma unroll
  for (int mi = 0; mi < 2; ++mi) {
    #pragma unroll
    for (int ni = 0; ni < 2; ++ni) {
      int n = bn + wn * 32 + ni * 16 + (lane & 15);
      float bv = bias[n];
      #pragma unroll
      for (int v = 0; v < 8; ++v) {
        int m = bm + wm * 32 + mi * 16 + v + ((lane >> 4) << 3);
        if (m < M) C[(size_t)m * HM + n] = c[mi][ni][v] + bv;
      }
    }
  }
}

// ---------------- row normalize (mean / std ddof=1, nan->0) ----------------
__global__ __launch_bounds__(256) void rownorm_kernel(
    const float* __restrict__ h, float* __restrict__ H0, float* __restrict__ out)
{
  __shared__ float sdata[256];
  size_t idx = (size_t)blockIdx.x * HM + threadIdx.x;
  float v = h[idx];
  float mean = block_reduce_sum(v, sdata) * (1.0f / HM);
  float d = v - mean;
  float var = block_reduce_sum(d * d, sdata) * (1.0f / (HM - 1));
  float stdv = sqrtf(var);
  float o = d / stdv;
  if (o != o) o = 0.0f;     // NaN -> 0
  H0[idx] = o;
  out[idx] = o;
}

// ---------------- generic zero (gated) ----------------
__global__ void zero_kernel(const float* gate, float* __restrict__ p, size_t n) {
  if (gate && gate[0] == 0.0f) return;
  size_t i = (size_t)blockIdx.x * blockDim.x + threadIdx.x;
  size_t st = (size_t)gridDim.x * blockDim.x;
  for (; i < n; i += st) p[i] = 0.0f;
}

// ---------------- degree count + edge expansion ----------------
__global__ __launch_bounds__(256) void deg_expand_kernel(
    const int* __restrict__ triples, float* __restrict__ deg,
    int* __restrict__ srcx, int* __restrict__ relx, int* __restrict__ dstx)
{
  int e = blockIdx.x * blockDim.x + threadIdx.x;
  if (e >= NE) return;
  int s = triples[3 * e + 0];
  int r = triples[3 * e + 1];
  int d = triples[3 * e + 2];
  srcx[e] = s;      relx[e] = r;      dstx[e] = d;
  srcx[NE + e] = d; relx[NE + e] = r; dstx[NE + e] = s;
  atomicAdd(&deg[(size_t)r * NT + s], 1.0f);
  atomicAdd(&deg[(size_t)r * NT + d], 1.0f);
}

__global__ __launch_bounds__(256) void edge_coef_kernel(
    const float* __restrict__ deg,
    const int* __restrict__ srcx, const int* __restrict__ relx, const int* __restrict__ dstx,
    float* __restrict__ invdeg, float* __restrict__ lap)
{
  int e = blockIdx.x * blockDim.x + threadIdx.x;
  if (e >= NE2) return;
  int s = srcx[e], r = relx[e], d = dstx[e];
  float ds_ = deg[(size_t)r * NT + s];
  float dd  = deg[(size_t)r * NT + d];
  invdeg[e] = 1.0f / ds_;
  lap[e] = 1.0f / sqrtf(ds_ * dd);
}

// ---------------- state init : u = 1/R, active = 1 ----------------
__global__ void init_state_kernel(float* state) {
  if (threadIdx.x == 0) {
    for (int r = 0; r < R_REL; ++r) state[r] = 1.0f / R_REL;
    state[8] = 0.0f;   // c_l1tr (set later)
    state[9] = 1.0f;   // active
  }
}

// ---------------- tv: sum X*X (gated grid-stride reduction) ----------------
__global__ __launch_bounds__(256) void sumsq_all_kernel(
    const float* gate, const float* __restrict__ p, float* __restrict__ acc, size_t n)
{
  if (gate && gate[0] == 0.0f) return;
  __shared__ float sdata[256];
  size_t i = (size_t)blockIdx.x * blockDim.x + threadIdx.x;
  size_t st = (size_t)gridDim.x * blockDim.x;
  float s = 0.0f;
  for (; i < n; i += st) { float v = p[i]; s += v * v; }
  s = block_reduce_sum(s, sdata);
  if (threadIdx.x == 0) atomicAdd(acc, s);
}

// ---------------- tv: per-edge dot(out[s], out[d]) * lap -> wsum[rel] ----------------
__global__ __launch_bounds__(256) void tv_edge_kernel(
    const float* gate, const float* __restrict__ out,
    const int* __restrict__ srcx, const int* __restrict__ relx, const int* __restrict__ dstx,
    const float* __restrict__ lap, float* __restrict__ wsum)
{
  if (gate && gate[0] == 0.0f) return;
  int e = blockIdx.x * 8 + (threadIdx.x >> 5);
  if (e >= NE2) return;
  int lane = threadIdx.x & 31;
  const float* ps = out + (size_t)srcx[e] * HM;
  const float* pd = out + (size_t)dstx[e] * HM;
  float acc = 0.0f;
  #pragma unroll
  for (int i = 0; i < 8; ++i) acc += ps[lane + 32 * i] * pd[lane + 32 * i];
  acc = wave_reduce_sum(acc);
  if (lane == 0) atomicAdd(&wsum[relx[e]], acc * lap[e]);
}

__global__ void tv_finalize_kernel(const float* gate, const float* __restrict__ xsq,
                                   const float* __restrict__ wsum, float* __restrict__ w,
                                   float* state, int firstIter)
{
  if (gate && gate[0] == 0.0f) return;
  if (threadIdx.x == 0) {
    float x = xsq[0];
    float l1 = 0.0f;
    for (int r = 0; r < R_REL; ++r) {
      float wr = (x - wsum[r]) / (float)NT;
      w[r] = wr;
      l1 += fabsf(wr);
    }
    if (firstIter) state[8] = l1;    // c_l1tr
  }
}

// ---------------- u inner loop (mirror descent, R=8, K_INNER=30) ----------------
__global__ void u_update_kernel(const float* __restrict__ w, float* state) {
  if (threadIdx.x != 0) return;
  float active = state[9];
  float u[R_REL];
  for (int r = 0; r < R_REL; ++r) u[r] = state[r];
  float l1 = 0.0f;
  for (int r = 0; r < R_REL; ++r) l1 += fabsf(w[r]);
  const float ba = 0.2f;                       // 2*BETA/ALPHA
  const float logR2 = 2.0f * logf((float)R_REL);
  float fi = l1 + ba;
  if (active != 0.0f) {
    bool act = true;
    for (int t = 11; t < 11 + 30 && act; ++t) {
      float Tt = sqrtf(logR2 / ((float)t * fi * fi));
      float ut[R_REL];
      float s = 0.0f;
      for (int r = 0; r < R_REL; ++r) {
        ut[r] = u[r] * expf(-Tt * (ba * u[r] + w[r]));
        s += ut[r];
      }
      float d2 = 0.0f;
      for (int r = 0; r < R_REL; ++r) {
        float un = ut[r] / s;
        float df = un - u[r];
        d2 += df * df;
        u[r] = un;
      }
      act = sqrtf(d2) >= 0.0001f;              // THRESHOLD_C
    }
    for (int r = 0; r < R_REL; ++r) state[r] = u[r];
  }
  float update = (active != 0.0f && (l1 / state[8] >= 0.9f)) ? 1.0f : 0.0f;
  state[9] = update;                            // gates rest of iter + next iter
}

// ---------------- afw scatter: afw[src] += invdeg * u[rel] * out[dst] ----------------
__global__ __launch_bounds__(256) void afw_scatter_kernel(
    const float* gate, const float* __restrict__ out,
    const int* __restrict__ srcx, const int* __restrict__ relx, const int* __restrict__ dstx,
    const float* __restrict__ invdeg, const float* __restrict__ state,
    float* __restrict__ afw)
{
  if (gate[0] == 0.0f) return;
  int e = blockIdx.x * 8 + (threadIdx.x >> 5);
  if (e >= NE2) return;
  int lane = threadIdx.x & 31;
  float coef = invdeg[e] * state[relx[e]];
  const float* pd = out + (size_t)dstx[e] * HM;
  float* pa = afw + (size_t)srcx[e] * HM;
  #pragma unroll
  for (int i = 0; i < 8; ++i)
    atomicAdd(&pa[lane + 32 * i], coef * pd[lane + 32 * i]);
}

__global__ void out_update_kernel(const float* gate, const float* __restrict__ H0,
                                  const float* __restrict__ afw, float* __restrict__ out, size_t n)
{
  if (gate[0] == 0.0f) return;
  const float c0 = 1.0f / 11.0f;               // 1/(1+ALPHA)
  const float c1 = 10.0f / 11.0f;              // ALPHA/(1+ALPHA)
  size_t i = (size_t)blockIdx.x * blockDim.x + threadIdx.x;
  size_t st = (size_t)gridDim.x * blockDim.x;
  for (; i < n; i += st) out[i] = c0 * H0[i] + c1 * afw[i];
}

// ---------------- head: column sums, centered sumsq, logits + embeddings ----------------
__global__ __launch_bounds__(256) void colsum_kernel(
    const float* __restrict__ out, float* __restrict__ colsum)
{
  int r0 = blockIdx.x * 32;
  float s = 0.0f;
  for (int r = r0; r < r0 + 32 && r < NT; ++r)
    s += out[(size_t)r * HM + threadIdx.x];
  atomicAdd(&colsum[threadIdx.x], s);
}

__global__ __launch_bounds__(256) void center_sumsq_kernel(
    const float* __restrict__ out, const float* __restrict__ colsum, float* __restrict__ ssq)
{
  __shared__ float sdata[256];
  const float invN = 1.0f / (float)NT;
  size_t n = NTH;
  float s = 0.0f;
  for (size_t i = (size_t)blockIdx.x * 256 + threadIdx.x; i < n; i += (size_t)gridDim.x * 256) {
    int c = (int)(i & (HM - 1));
    float xp = out[i] - colsum[c] * invN;
    s += xp * xp;
  }
  s = block_reduce_sum(s, sdata);
  if (threadIdx.x == 0) atomicAdd(ssq, s);
}

__global__ __launch_bounds__(256) void head_kernel(
    const float* __restrict__ out, const float* __restrict__ colsum, const float* __restrict__ ssq,
    const float* __restrict__ Wout, const float* __restrict__ bout,
    float* __restrict__ logits, float* __restrict__ emb)
{
  __shared__ float xp[HM];
  int n = blockIdx.x, t = threadIdx.x;
  const float invN = 1.0f / (float)NT;
  float inv_norm = 1.0f / sqrtf(1e-6f + ssq[0] * invN);
  float v = out[(size_t)n * HM + t];
  emb[(size_t)n * HM + t] = v;                 // embeddings = out_f (uncentered)
  xp[t] = (v - colsum[t] * invN) * inv_norm;
  __syncthreads();
  if (t < NUM_CLASSES) {
    float s = bout[t];
    for (int c = 0; c < HM; ++c) s += xp[c] * Wout[c * NUM_CLASSES + t];
    logits[(size_t)n * NUM_CLASSES + t] = s;
  }
}

// ---------------- launch ----------------
extern "C" void kernel_launch(void* const* d_in, const int* in_sizes, int n_in,
                              void* d_out, int out_size, void* d_ws, size_t ws_size,
                              hipStream_t stream) {
  const float* x     = (const float*)d_in[0];
  const float* gene  = (const float*)d_in[1];
  const int*   trip  = (const int*)  d_in[2];
  const float* W0    = (const float*)d_in[3];
  const float* b0    = (const float*)d_in[4];
  const float* W1    = (const float*)d_in[5];
  const float* b1    = (const float*)d_in[6];
  const float* Wout  = (const float*)d_in[7];
  const float* bout  = (const float*)d_in[8];

  // workspace layout (floats)
  float* ws    = (float*)d_ws;
  float* H0    = ws;                               // NTH
  float* OUT   = H0 + NTH;                         // NTH
  float* AFW   = OUT + NTH;                        // NTH (also temp for raw h)
  float* DEG   = AFW + NTH;                        // R*NT
  float* INV   = DEG + (size_t)R_REL * NT;         // NE2
  float* LAP   = INV + NE2;                        // NE2
  int*   SRCX  = (int*)(LAP + NE2);                // NE2
  int*   RELX  = SRCX + NE2;                       // NE2
  int*   DSTX  = RELX + NE2;                       // NE2
  float* WSUM  = (float*)(DSTX + NE2);             // 8
  float* XSQ   = WSUM + R_REL;                     // 1  (adjacent to WSUM: zero 9 at once)
  float* WVEC  = XSQ + 1;                          // 8
  float* COLS  = WVEC + R_REL;                     // 256
  float* SSQ   = COLS + HM;                        // 1  (adjacent: zero 257 at once)
  float* STATE = SSQ + 1;                          // 16: u[8], c_l1tr, active
  const float* GATE = STATE + 9;

  float* LOGITS = (float*)d_out;
  float* EMB    = LOGITS + (size_t)NT * NUM_CLASSES;

  // 1) WMMA GEMMs: raw h into AFW temp buffer
  gemm_bias_kernel<<<dim3((N_X + BLK_M - 1) / BLK_M, HM / BLK_N), 256, 0, stream>>>(
      x, LDX, ACOL, N_X, KX, W0, b0, AFW);
  gemm_bias_kernel<<<dim3((N_GENE + BLK_M - 1) / BLK_M, HM / BLK_N), 256, 0, stream>>>(
      gene, N_GENE, 0, N_GENE, N_GENE, W1, b1, AFW + (size_t)N_X * HM);

  // 2) row normalize -> H0, OUT
  rownorm_kernel<<<NT, 256, 0, stream>>>(AFW, H0, OUT);

  // 3) degrees + expanded edges + per-edge coefficients
  zero_kernel<<<2048, 256, 0, stream>>>(nullptr, DEG, (size_t)R_REL * NT);
  deg_expand_kernel<<<(NE + 255) / 256, 256, 0, stream>>>(trip, DEG, SRCX, RELX, DSTX);
  edge_coef_kernel<<<(NE2 + 255) / 256, 256, 0, stream>>>(DEG, SRCX, RELX, DSTX, INV, LAP);

  // 4) state init
  init_state_kernel<<<1, 32, 0, stream>>>(STATE);

  // 5) K_OUTER = 4 outer iterations (unrolled; all gating is device-side)
  for (int it = 0; it < 4; ++it) {
    zero_kernel<<<1, 32, 0, stream>>>(GATE, WSUM, R_REL + 1);                     // wsum + xsq
    sumsq_all_kernel<<<1024, 256, 0, stream>>>(GATE, OUT, XSQ, NTH);
    tv_edge_kernel<<<(NE2 + 7) / 8, 256, 0, stream>>>(GATE, OUT, SRCX, RELX, DSTX, LAP, WSUM);
    tv_finalize_kernel<<<1, 32, 0, stream>>>(GATE, XSQ, WSUM, WVEC, STATE, it == 0 ? 1 : 0);
    u_update_kernel<<<1, 32, 0, stream>>>(WVEC, STATE);                           // sets GATE=update
    zero_kernel<<<2048, 256, 0, stream>>>(GATE, AFW, NTH);
    afw_scatter_kernel<<<(NE2 + 7) / 8, 256, 0, stream>>>(GATE, OUT, SRCX, RELX, DSTX,
                                                          INV, STATE, AFW);
    out_update_kernel<<<2048, 256, 0, stream>>>(GATE, H0, AFW, OUT, NTH);
  }

  // 6) head
  zero_kernel<<<2, 256, 0, stream>>>(nullptr, COLS, HM + 1);                      // colsum + ssq
  colsum_kernel<<<(NT + 31) / 32, 256, 0, stream>>>(OUT, COLS);
  center_sumsq_kernel<<<1024, 256, 0, stream>>>(OUT, COLS, SSQ);
  head_kernel<<<NT, 256, 0, stream>>>(OUT, COLS, SSQ, Wout, bout, LOGITS, EMB);

  (void)in_sizes; (void)n_in; (void)out_size; (void)ws_size;
}